// nearest_proj_layer_47081431498925
// MI455X (gfx1250) — compile-verified
//
#include <hip/hip_runtime.h>
#include <stdint.h>

// ---------------------------------------------------------------------------
// nearest_proj_layer for MI455X (gfx1250)
//   out[b,t] = sum_k x[b, idx_k(b,t)] * proj[k],  idx = 16 smallest of d_mat[b,t,:]
//
// Streaming-selection kernel: 1 GiB of d_mat read exactly once -> bandwidth
// bound (~46us floor @ 23.3 TB/s). No matmul structure (16-elem dot -> one
// scalar per row), so WMMA would waste 15/16 of a tile and add sync; the
// CDNA5 feature that matters is the async global->LDS data mover:
//   - GLOBAL_LOAD_ASYNC_TO_LDS_B128, th:TH_LOAD_NT (1 GiB >> 192 MB L2,
//     every line consumed once)
//   - 4-deep ring buffer, 3 chunks (12 async ops, 6 KB) in flight per wave,
//     staged s_wait_asynccnt 12/8/4/0 (async loads complete in order)
// Selection: packed (f32bits<<32 | index) u64 keys reproduce lax.top_k's
// ascending order + lowest-index tie-break exactly.
// ---------------------------------------------------------------------------

namespace {
constexpr int kB = 8;
constexpr int kS = 8192;        // row length (source positions)
constexpr int kT = 4096;        // targets per batch
constexpr int kK = 16;          // top-k
constexpr int kWaves = 8;       // waves per block
constexpr int kChunk = 512;     // floats staged per chunk per wave (2 KB)
constexpr int kNChunk = kS / kChunk;  // 16
constexpr int kRing = 4;        // ring-buffer depth (chunks in flight: 3)
}

typedef unsigned long long u64;
typedef unsigned int u32;

// Async DMA: 16 bytes/lane, global -> LDS. LDS dest address from a VGPR
// (byte offset within LDS), global address from a VGPR pair. Tracked by
// ASYNCcnt; completes in order with other async loads from this wave.
__device__ __forceinline__ void async_copy_b128(u32 lds_off, u64 gaddr) {
  asm volatile("global_load_async_to_lds_b128 %0, %1, off th:TH_LOAD_NT"
               :: "v"(lds_off), "v"(gaddr)
               : "memory");
}
__device__ __forceinline__ void wait_async_le12() {
  asm volatile("s_wait_asynccnt 0xc" ::: "memory");
}
__device__ __forceinline__ void wait_async_le8() {
  asm volatile("s_wait_asynccnt 0x8" ::: "memory");
}
__device__ __forceinline__ void wait_async_le4() {
  asm volatile("s_wait_asynccnt 0x4" ::: "memory");
}
__device__ __forceinline__ void wait_async_0() {
  asm volatile("s_wait_asynccnt 0x0" ::: "memory");
}

// Branch-free insertion of key t into ascending sorted 16-entry list,
// dropping the old maximum. Keys are unique ((value,index) packed).
__device__ __forceinline__ void insert16(u64 (&lst)[kK], u64 t) {
  if (t < lst[kK - 1]) {
#pragma unroll
    for (int p = 0; p < kK; ++p) {
      const bool lt = lst[p] < t;
      const u64 lo = lt ? lst[p] : t;
      const u64 hi = lt ? t : lst[p];
      lst[p] = lo;
      t = hi;
    }
  }
}

__global__ __launch_bounds__(32 * kWaves)
void nearest_proj_topk_kernel(const float* __restrict__ x,
                              const float* __restrict__ dmat,
                              const float* __restrict__ proj,
                              float* __restrict__ out) {
  // Per-wave 4-deep ring: 4 x 2 KB, 64 KB per block total (5 blocks/WGP).
  __shared__ __align__(16) float lbuf[kWaves][kRing][kChunk];

  const int lane = threadIdx.x & 31;
  const int wave = threadIdx.x >> 5;
  const int row = blockIdx.x * kWaves + wave;   // flat (b,t), 0..32767
  if (row >= kB * kT) return;
  const int bIdx = row / kT;

  const float* rowp = dmat + (size_t)row * kS;
  u32 ldsBase[kRing];
#pragma unroll
  for (int r = 0; r < kRing; ++r)
    ldsBase[r] = (u32)(uintptr_t)(&lbuf[wave][r][0]) + (u32)(lane * 16);

  // Per-lane ascending top-16 keys: (f32 bits << 32) | source index.
  u64 lst[kK];
#pragma unroll
  for (int i = 0; i < kK; ++i) lst[i] = ~0ull;

  // Prime the pipeline: chunks 0..2 into ring slots 0..2 (12 async b128s).
#pragma unroll
  for (int c = 0; c < kRing - 1; ++c) {
    const u64 g = (u64)(uintptr_t)(rowp + c * kChunk) + (u64)(lane * 16);
    const u32 l = ldsBase[c];
#pragma unroll
    for (int i = 0; i < 4; ++i) async_copy_b128(l + i * 512, g + i * 512);
  }

  for (int c = 0; c < kNChunk; ++c) {
    const int cur = c & (kRing - 1);
    if (c + 3 < kNChunk) {
      // Issue chunk c+3 into the slot vacated by chunk c-1 (its LDS reads
      // were consumed last iteration), then wait so chunk c is resident:
      // up to 16 async ops pending, newest 12 belong to chunks c+1..c+3.
      const u64 g = (u64)(uintptr_t)(rowp + (c + 3) * kChunk) + (u64)(lane * 16);
      const u32 l = ldsBase[(c + 3) & (kRing - 1)];
#pragma unroll
      for (int i = 0; i < 4; ++i) async_copy_b128(l + i * 512, g + i * 512);
      wait_async_le12();
    } else if (c + 2 < kNChunk) {
      wait_async_le8();      // pending: chunks c..c+2 -> <=8 leaves c done
    } else if (c + 1 < kNChunk) {
      wait_async_le4();      // pending: chunks c..c+1 -> <=4 leaves c done
    } else {
      wait_async_0();        // last chunk
    }

    const float* bp = &lbuf[wave][cur][0];
    for (int i = 0; i < 4; ++i) {           // 4 x float4 per lane per chunk
      const int off = i * 128 + lane * 4;
      const float4 v = *(const float4*)(bp + off);
      const u32 s0 = (u32)(c * kChunk + off);
      const u64 k0 = ((u64)__float_as_uint(v.x) << 32) | (u64)(s0 + 0);
      const u64 k1 = ((u64)__float_as_uint(v.y) << 32) | (u64)(s0 + 1);
      const u64 k2 = ((u64)__float_as_uint(v.z) << 32) | (u64)(s0 + 2);
      const u64 k3 = ((u64)__float_as_uint(v.w) << 32) | (u64)(s0 + 3);
      const u64 m01 = k0 < k1 ? k0 : k1;
      const u64 m23 = k2 < k3 ? k2 : k3;
      const u64 gm = m01 < m23 ? m01 : m23;
      if (gm < lst[kK - 1]) {               // rare: some candidate qualifies
        insert16(lst, k0);
        insert16(lst, k1);
        insert16(lst, k2);
        insert16(lst, k3);
      }
    }
  }

  // Merge 32 sorted lists -> global ascending top-16 (16 rounds of wave-min
  // + pop on the unique winning lane; keys are unique so equality picks one).
  u64 sel = ~0ull;
  for (int r = 0; r < kK; ++r) {
    const u64 head = lst[0];
    u64 m = head;
#pragma unroll
    for (int o = 16; o > 0; o >>= 1) {
      const u64 other = (u64)__shfl_xor((unsigned long long)m, o, 32);
      m = other < m ? other : m;
    }
    if (lane == r) sel = m;
    if (head == m) {                        // pop winner's head
#pragma unroll
      for (int p = 0; p < kK - 1; ++p) lst[p] = lst[p + 1];
      lst[kK - 1] = ~0ull;
    }
  }

  // Lane r holds the r-th smallest (ascending, lax.top_k order): gather x,
  // scale by proj[r], wave-sum, lane 0 writes the scalar.
  float val = 0.0f;
  if (lane < kK && sel != ~0ull) {
    const u32 sidx = (u32)(sel & 0xffffffffu);
    val = x[(size_t)bIdx * kS + sidx] * proj[lane];
  }
#pragma unroll
  for (int o = 16; o > 0; o >>= 1) val += __shfl_xor(val, o, 32);
  if (lane == 0) out[row] = val;
}

extern "C" void kernel_launch(void* const* d_in, const int* in_sizes, int n_in,
                              void* d_out, int out_size, void* d_ws, size_t ws_size,
                              hipStream_t stream) {
  (void)in_sizes; (void)n_in; (void)out_size; (void)d_ws; (void)ws_size;
  const float* x    = (const float*)d_in[0];   // [B, S, 1] f32
  const float* dmat = (const float*)d_in[1];   // [B, T, S] f32
  const float* proj = (const float*)d_in[2];   // [K, 1] f32
  float* out = (float*)d_out;                  // [B, T, 1] f32

  const int rows = kB * kT;                    // 32768
  const dim3 grid(rows / kWaves);              // 4096 blocks
  const dim3 block(32 * kWaves);               // 256 threads = 8 waves
  nearest_proj_topk_kernel<<<grid, block, 0, stream>>>(x, dmat, proj, out);
}